// Better_Transformer_14448269984167
// MI455X (gfx1250) — compile-verified
//
#include <hip/hip_runtime.h>
#include <math.h>

#define BATCH   16384
#define IN_SIZE 4096
#define N_PART  32
#define INT_DIM 128
#define BN_EPS  1e-5f

#define ROWS      64     // batch rows per block
#define KQ        32     // K-quarter staged per LDS pass
#define XN_STRIDE 132    // 128 + 4 pad -> stride % 64 == 4 (conflict-free A reads)
#define WT_STRIDE 36     // 32 + 4 pad  -> 4*odd stride, 16 distinct banks for B reads

typedef float v2f __attribute__((ext_vector_type(2)));
typedef float v8f __attribute__((ext_vector_type(8)));

// ---------------- kernel 0: zero workspace accumulators ----------------
__global__ __launch_bounds__(256) void bt_zero(float* __restrict__ ws) {
    ws[blockIdx.x * 256 + threadIdx.x] = 0.0f;
}

// ---------------- kernel 1: per-feature sum / sum-of-squares ----------------
__global__ __launch_bounds__(256) void bt_stats(const float* __restrict__ x,
                                                float* __restrict__ ws) {
    const int f  = blockIdx.x * 256 + threadIdx.x;   // feature
    const int r0 = blockIdx.y * 256;                 // row chunk
    const float* p = x + (size_t)r0 * IN_SIZE + f;
    float s = 0.0f, q = 0.0f;
    for (int i = 0; i < 256; ++i) {
        float v = p[(size_t)i * IN_SIZE];
        s += v;
        q = fmaf(v, v, q);
    }
    atomicAdd(&ws[f], s);
    atomicAdd(&ws[IN_SIZE + f], q);
}

// ---------------- kernel 2: fold BN into scale/shift ----------------
__global__ __launch_bounds__(256) void bt_finalize(float* __restrict__ ws,
                                                   const float* __restrict__ gamma,
                                                   const float* __restrict__ beta) {
    const int f = blockIdx.x * 256 + threadIdx.x;
    const float inv_b = 1.0f / (float)BATCH;
    float mean = ws[f] * inv_b;
    float var  = fmaf(-mean, mean, ws[IN_SIZE + f] * inv_b);   // biased variance
    float sc   = gamma[f] * rsqrtf(var + BN_EPS);
    ws[2 * IN_SIZE + f] = sc;
    ws[3 * IN_SIZE + f] = fmaf(-mean, sc, beta[f]);
}

// ---------------- kernel 3: BN-apply + block GEMM (WMMA) + tanh + residual ----------------
__global__ __launch_bounds__(256) void bt_main(const float* __restrict__ x,
                                               const float* __restrict__ w,
                                               const float* __restrict__ bias,
                                               const float* __restrict__ ws,
                                               float* __restrict__ out) {
    __shared__ float Wt[INT_DIM * WT_STRIDE];   // W[p] K-quarter, transposed: Wt[n][k]
    __shared__ float Xn[ROWS * XN_STRIDE];      // BN-applied x tile

    const int p    = blockIdx.y;                // partition
    const int row0 = blockIdx.x * ROWS;
    const int c0   = p * INT_DIM;               // feature base for this partition
    const int tid  = threadIdx.x;

    const float* __restrict__ scale = ws + 2 * IN_SIZE;
    const float* __restrict__ shift = ws + 3 * IN_SIZE;
    const float* __restrict__ Wp    = w + (size_t)p * INT_DIM * INT_DIM;

    const int wave = tid >> 5;
    const int lane = tid & 31;
    const int lm   = lane & 15;                 // M (A) / N (B) index within tile
    const int lh   = lane >> 4;                 // lane half -> K offset pairs
    const int rt   = wave >> 1;                 // row tile 0..3
    const int ntb  = (wave & 1) * 4;            // base N tile (each wave owns 4)

    v8f acc[4];
#pragma unroll
    for (int j = 0; j < 4; ++j) acc[j] = (v8f){0, 0, 0, 0, 0, 0, 0, 0};

    for (int s = 0; s < 4; ++s) {               // four K-quarters of 32
        // stage W[p][s*32 .. s*32+31][:] transposed into LDS (coalesced global reads)
        for (int i = tid; i < KQ * INT_DIM; i += 256) {
            const int kl = i >> 7;              // local K  0..31
            const int n  = i & 127;             // N column 0..127
            Wt[n * WT_STRIDE + kl] = Wp[(size_t)(s * KQ + kl) * INT_DIM + n];
        }
        if (s == 0) {
            // stage x tile with BN applied
            for (int i = tid; i < ROWS * INT_DIM; i += 256) {
                const int r = i >> 7;
                const int c = i & 127;
                float v = x[(size_t)(row0 + r) * IN_SIZE + c0 + c];
                Xn[r * XN_STRIDE + c] = fmaf(v, scale[c0 + c], shift[c0 + c]);
            }
        }
        __syncthreads();

        const float* ap = Xn + (rt * 16 + lm) * XN_STRIDE + s * KQ + 2 * lh;
        const float* bp = Wt + (ntb * 16 + lm) * WT_STRIDE + 2 * lh;

#pragma unroll
        for (int kk = 0; kk < KQ; kk += 4) {
            v2f a = *(const v2f*)(ap + kk);
#pragma unroll
            for (int j = 0; j < 4; ++j) {
                v2f b = *(const v2f*)(bp + j * 16 * WT_STRIDE + kk);
                acc[j] = __builtin_amdgcn_wmma_f32_16x16x4_f32(false, a, false, b,
                                                               (short)0, acc[j],
                                                               false, false);
            }
        }
        __syncthreads();                        // protect Wt before next stage load
    }

    // epilogue: tanh(y + bias) + x  (C/D layout: vgpr e, lane half -> row = 8*lh + e)
#pragma unroll
    for (int j = 0; j < 4; ++j) {
        const int gcol = c0 + (ntb + j) * 16 + lm;
        const float bv = bias[gcol];
#pragma unroll
        for (int e = 0; e < 8; ++e) {
            const int grow = row0 + rt * 16 + lh * 8 + e;
            const size_t off = (size_t)grow * IN_SIZE + gcol;
            out[off] = tanhf(acc[j][e] + bv) + x[off];
        }
    }
}

extern "C" void kernel_launch(void* const* d_in, const int* in_sizes, int n_in,
                              void* d_out, int out_size, void* d_ws, size_t ws_size,
                              hipStream_t stream) {
    const float* x     = (const float*)d_in[0];
    const float* wts   = (const float*)d_in[1];
    const float* bias  = (const float*)d_in[2];
    const float* gamma = (const float*)d_in[3];
    const float* beta  = (const float*)d_in[4];
    float*       out   = (float*)d_out;
    float*       ws    = (float*)d_ws;   // [sum | sumsq | scale | shift] x 4096

    bt_zero<<<(2 * IN_SIZE) / 256, 256, 0, stream>>>(ws);

    dim3 gs(IN_SIZE / 256, BATCH / 256);
    bt_stats<<<gs, 256, 0, stream>>>(x, ws);

    bt_finalize<<<IN_SIZE / 256, 256, 0, stream>>>(ws, gamma, beta);

    dim3 gm(BATCH / ROWS, N_PART);
    bt_main<<<gm, 256, 0, stream>>>(x, wts, bias, ws, out);
}